// BidirectionalALiBi_26980984553599
// MI455X (gfx1250) — compile-verified
//
#include <hip/hip_runtime.h>
#include <math.h>

// out[h,i,j] = |j-i| * coef(h,i,j)
//   coef = alpha[h] if (i==0 || j==0), gamma[h] if j>i, beta[h] if j<i
//   (j==i contributes 0 since dist==0)
//
// Pure store-bandwidth kernel: 268 MB written once, ~0 read.
// Strategy: b128 non-temporal stores, block-uniform (h,i) so coefficient
// loads and row addressing stay scalar (SGPR), no integer divides anywhere.

typedef float v4f __attribute__((ext_vector_type(4)));

__global__ __launch_bounds__(256) void BidirectionalALiBi_kernel(
    const float* __restrict__ alpha,
    const float* __restrict__ beta,
    const float* __restrict__ gamma,
    float* __restrict__ out,
    int S)
{
    const int j0 = (blockIdx.x * 256 + threadIdx.x) * 4;  // first of 4 columns
    if (j0 >= S) return;

    const int i = blockIdx.y;   // row   (block-uniform -> SGPR)
    const int h = blockIdx.z;   // head  (block-uniform -> SGPR)

    // Scalar (wave-uniform) coefficient loads -> s_load path.
    const float a = alpha[h];
    const float b = beta[h];
    const float g = gamma[h];

    const size_t base = ((size_t)h * (size_t)S + (size_t)i) * (size_t)S;

    if (j0 + 3 < S) {
        v4f v;
#pragma unroll
        for (int k = 0; k < 4; ++k) {
            const int j = j0 + k;
            const int d = j - i;
            const float dist = fabsf((float)d);
            float coef = (d > 0) ? g : b;          // j>i -> gamma, j<i -> beta
            if ((i == 0) | (j == 0)) coef = a;     // edge row/column -> alpha
            v[k] = dist * coef;
        }
        // Write-once 256MB stream (> 192MB L2): bypass with NT hint,
        // lowers to global_store_b128 ... th:TH_STORE_NT.
        __builtin_nontemporal_store(v, (v4f*)(out + base + (size_t)j0));
    } else {
        // Tail path (not taken for S % 4 == 0, e.g. S = 2048).
        for (int j = j0; j < S; ++j) {
            const int d = j - i;
            const float dist = fabsf((float)d);
            float coef = (d > 0) ? g : b;
            if ((i == 0) | (j == 0)) coef = a;
            __builtin_nontemporal_store(dist * coef, out + base + (size_t)j);
        }
    }
}

extern "C" void kernel_launch(void* const* d_in, const int* in_sizes, int n_in,
                              void* d_out, int out_size, void* d_ws, size_t ws_size,
                              hipStream_t stream) {
    (void)n_in; (void)d_ws; (void)ws_size;

    const float* alpha = (const float*)d_in[0];
    const float* beta  = (const float*)d_in[1];
    const float* gamma = (const float*)d_in[2];
    float* out = (float*)d_out;

    const int H = in_sizes[0];                       // 16
    // seq_len lives on-device (unreadable under graph capture); recover it
    // from out_size = H * S * S deterministically on the host.
    const long long ss = (long long)out_size / (long long)(H > 0 ? H : 1);
    const int S = (int)llround(sqrt((double)ss));    // 2048

    const int vec_per_row = (S + 3) / 4;             // float4 slots per row
    dim3 block(256);
    dim3 grid((vec_per_row + 255) / 256, S, H);      // (2, 2048, 16) for S=2048

    BidirectionalALiBi_kernel<<<grid, block, 0, stream>>>(alpha, beta, gamma, out, S);
}